// BredictNetwork_68229850464587
// MI455X (gfx1250) — compile-verified
//
#include <hip/hip_runtime.h>
#include <hip/hip_bf16.h>

#define NL   8
#define HIDD 4
#define TT   2048
#define BB   4096

typedef float v2f __attribute__((ext_vector_type(2)));
typedef float v8f __attribute__((ext_vector_type(8)));

// Compact tanh: prefer the gfx1250 TANH trans op; otherwise a 5-op
// formulation (mul, exp, add, rcp, fma) that saturates correctly at +-1.
#if defined(__has_builtin)
# if __has_builtin(__builtin_amdgcn_tanhf)
#  define TANHF(v) __builtin_amdgcn_tanhf(v)
# endif
#endif
#ifndef TANHF
__device__ __forceinline__ float fast_tanhf_(float x) {
    float e = __expf(2.0f * x);                       // v_mul + v_exp_f32
    return 1.0f - 2.0f * __builtin_amdgcn_rcpf(e + 1.0f);  // v_add + v_rcp + v_fma
}
# define TANHF(v) fast_tanhf_(v)
#endif

// One-instruction lane swap + add via ds_swizzle (group-of-32 XOR patterns).
#define SWIZ_ADD(v, pat)                                                     \
    (v) += __int_as_float(__builtin_amdgcn_ds_swizzle(__float_as_int(v), (pat)))

// ---------------------------------------------------------------------------
// Kernel 1: 8-layer tanh RNN, layer-per-wave systolic pipeline.
//   block = 256 threads = 8 waves; wave l handles layer l for 32 batch rows
//   (lane-per-batch). Activations flow wave(l) -> wave(l+1) through a
//   triple-buffered LDS ring; one workgroup barrier per superstep.
//   grid = BB/32 = 128 blocks -> 1024 waves on the device.
// ---------------------------------------------------------------------------
__global__ __launch_bounds__(256, 1) void rnn8_pipeline(
    const float* __restrict__ x,        // [B, T, 4]
    const float* __restrict__ hidden0,  // [8, B, 4] (zeros, but honor it)
    const float* __restrict__ Wih,      // [8, 4, 4]
    const float* __restrict__ Whh,      // [8, 4, 4]
    const float* __restrict__ bih,      // [8, 4]
    const float* __restrict__ bhh,      // [8, 4]
    float* __restrict__ hn)             // [8, B, 4]  (= d_out + B)
{
    __shared__ __align__(16) float buf[NL - 1][3][32 * HIDD];

    const int tid  = threadIdx.x;
    const int w    = __builtin_amdgcn_readfirstlane(tid >> 5);  // layer id, wave-uniform
    const int lane = tid & 31;
    const int b    = blockIdx.x * 32 + lane;                    // global batch row

    // Per-wave weights: addresses are wave-uniform -> scalar loads.
    float wih[16], whh[16], bs[4];
#pragma unroll
    for (int i = 0; i < 16; ++i) { wih[i] = Wih[w * 16 + i]; whh[i] = Whh[w * 16 + i]; }
#pragma unroll
    for (int j = 0; j < 4; ++j)  bs[j] = bih[w * 4 + j] + bhh[w * 4 + j];

    const float4 hv0 = *((const float4*)hidden0 + (size_t)w * BB + b);
    float h0 = hv0.x, h1 = hv0.y, h2 = hv0.z, h3 = hv0.w;

    const float4* xrow = (const float4*)x + (size_t)b * TT;  // contiguous per-lane stream

    int slot = 0;                                   // rotating t % 3 (no div/mod)
    for (int s = 0; s < TT + NL - 1; ++s) {
        const int  t   = s - w;
        const bool act = (t >= 0) && (t < TT);      // wave-uniform
        if (act) {
            float4 in;
            if (w == 0) {
                in = xrow[t];
                if (t + 32 < TT) __builtin_prefetch(&xrow[t + 32], 0, 3);
            } else {
                in = *(const float4*)&buf[w - 1][slot][lane * HIDD];
            }
            float a0 = bs[0], a1 = bs[1], a2 = bs[2], a3 = bs[3];
            a0 = fmaf(in.x, wih[0],  a0); a0 = fmaf(in.y, wih[1],  a0);
            a0 = fmaf(in.z, wih[2],  a0); a0 = fmaf(in.w, wih[3],  a0);
            a0 = fmaf(h0,   whh[0],  a0); a0 = fmaf(h1,   whh[1],  a0);
            a0 = fmaf(h2,   whh[2],  a0); a0 = fmaf(h3,   whh[3],  a0);

            a1 = fmaf(in.x, wih[4],  a1); a1 = fmaf(in.y, wih[5],  a1);
            a1 = fmaf(in.z, wih[6],  a1); a1 = fmaf(in.w, wih[7],  a1);
            a1 = fmaf(h0,   whh[4],  a1); a1 = fmaf(h1,   whh[5],  a1);
            a1 = fmaf(h2,   whh[6],  a1); a1 = fmaf(h3,   whh[7],  a1);

            a2 = fmaf(in.x, wih[8],  a2); a2 = fmaf(in.y, wih[9],  a2);
            a2 = fmaf(in.z, wih[10], a2); a2 = fmaf(in.w, wih[11], a2);
            a2 = fmaf(h0,   whh[8],  a2); a2 = fmaf(h1,   whh[9],  a2);
            a2 = fmaf(h2,   whh[10], a2); a2 = fmaf(h3,   whh[11], a2);

            a3 = fmaf(in.x, wih[12], a3); a3 = fmaf(in.y, wih[13], a3);
            a3 = fmaf(in.z, wih[14], a3); a3 = fmaf(in.w, wih[15], a3);
            a3 = fmaf(h0,   whh[12], a3); a3 = fmaf(h1,   whh[13], a3);
            a3 = fmaf(h2,   whh[14], a3); a3 = fmaf(h3,   whh[15], a3);

            h0 = TANHF(a0); h1 = TANHF(a1); h2 = TANHF(a2); h3 = TANHF(a3);

            if (w < NL - 1)
                *(float4*)&buf[w][slot][lane * HIDD] = make_float4(h0, h1, h2, h3);
            slot = (slot == 2) ? 0 : slot + 1;      // advance only on active steps
        }
        __syncthreads();
    }

    // Final hidden per layer: h_n[l][b][:]
    *((float4*)hn + (size_t)w * BB + b) = make_float4(h0, h1, h2, h3);
}

// ---------------------------------------------------------------------------
// Kernel 2: FC head with f32 WMMA.
//   relu(h_n[7] @ Wfc^T + bfc) @ Wfc2^T + bfc2   for 4096 rows.
//   One wave per 16-row tile: A = 16x4 f32 tile (2 VGPRs), 4 WMMAs cover the
//   64 output columns of FC1 (B = 4x16 chunk of Wfc^T, C = bias broadcast),
//   FC2 fused as per-lane FMA + ds_swizzle butterfly over the 16-lane halves.
// ---------------------------------------------------------------------------
__global__ __launch_bounds__(128, 1) void fc_head_wmma(
    const float* __restrict__ hn7,   // [B, 4]
    const float* __restrict__ Wfc,   // [64, 4]
    const float* __restrict__ bfc,   // [64]
    const float* __restrict__ Wfc2,  // [1, 64]
    const float* __restrict__ bfc2,  // [1]
    float* __restrict__ out)         // [B]
{
    const int lane = threadIdx.x & 31;
    const int wv   = threadIdx.x >> 5;
    const int tile = blockIdx.x * 4 + wv;
    const int row0 = tile * 16;
    const int m    = lane & 15;          // row (for A) / column (for B,C,D)
    const int kb   = (lane >> 4) << 1;   // K pair: lanes 0-15 -> K=0,1 ; 16-31 -> K=2,3

    // A matrix: 16x4 f32 (VGPR0 = K even, VGPR1 = K odd; hi lanes = K+2)
    v2f a;
    a.x = hn7[(row0 + m) * 4 + kb];
    a.y = hn7[(row0 + m) * 4 + kb + 1];

    float acc[8];
#pragma unroll
    for (int r = 0; r < 8; ++r) acc[r] = 0.f;

#pragma unroll
    for (int c4 = 0; c4 < 4; ++c4) {
        const int n = c4 * 16 + m;       // FC1 output column for this lane
        v2f bm;                          // B matrix: 4x16 chunk of Wfc^T
        bm.x = Wfc[n * 4 + kb];
        bm.y = Wfc[n * 4 + kb + 1];
        const float bias = bfc[n];
        v8f cc;
#pragma unroll
        for (int r = 0; r < 8; ++r) cc[r] = bias;   // C[M][n] = bfc[n] for all M

        v8f d = __builtin_amdgcn_wmma_f32_16x16x4_f32(
            /*neg_a=*/false, a, /*neg_b=*/false, bm,
            /*c_mod=*/(short)0, cc, /*reuse_a=*/false, /*reuse_b=*/false);

        const float w2 = Wfc2[n];        // FC2 weight for this column
#pragma unroll
        for (int r = 0; r < 8; ++r) acc[r] += fmaxf(d[r], 0.f) * w2;
    }

    const float b2 = bfc2[0];
    // D layout: VGPR r, lanes 0-15 -> row r ; lanes 16-31 -> row r+8.
    // Reduce each row across its 16-lane half; XOR swizzles 1,2,4,8 stay in-half.
#pragma unroll
    for (int r = 0; r < 8; ++r) {
        float v = acc[r];
        SWIZ_ADD(v, 0x041F);   // SWAPX1
        SWIZ_ADD(v, 0x081F);   // SWAPX2
        SWIZ_ADD(v, 0x101F);   // SWAPX4
        SWIZ_ADD(v, 0x201F);   // SWAPX8
        if (m == 0)
            out[row0 + (lane >> 4) * 8 + r] = v + b2;
    }
}

// ---------------------------------------------------------------------------
extern "C" void kernel_launch(void* const* d_in, const int* in_sizes, int n_in,
                              void* d_out, int out_size, void* d_ws, size_t ws_size,
                              hipStream_t stream) {
    (void)in_sizes; (void)n_in; (void)out_size; (void)d_ws; (void)ws_size;

    const float* x      = (const float*)d_in[0];
    const float* hidden = (const float*)d_in[1];
    const float* Wih    = (const float*)d_in[2];
    const float* Whh    = (const float*)d_in[3];
    const float* bihp   = (const float*)d_in[4];
    const float* bhhp   = (const float*)d_in[5];
    const float* Wfc    = (const float*)d_in[6];
    const float* bfc    = (const float*)d_in[7];
    const float* Wfc2   = (const float*)d_in[8];
    const float* bfc2   = (const float*)d_in[9];

    float* out = (float*)d_out;      // output [B,1] = first 4096 floats
    float* hn  = out + BB;           // h_n [8,B,4] follows

    rnn8_pipeline<<<BB / 32, 256, 0, stream>>>(x, hidden, Wih, Whh, bihp, bhhp, hn);

    const float* hn7 = hn + (size_t)7 * BB * 4;  // layer-7 last hidden == xs[-1]
    fc_head_wmma<<<BB / 64, 128, 0, stream>>>(hn7, Wfc, bfc, Wfc2, bfc2, out);
}